// ISNNet_2216203125034
// MI455X (gfx1250) — compile-verified
//
#include <hip/hip_runtime.h>

// CDNA5 / gfx1250, wave32. fp32 WMMA 16x16x4 throughout (precision demands fp32
// for a 2048-step contractive recurrence).

typedef float v2f __attribute__((ext_vector_type(2)));
typedef float v8f __attribute__((ext_vector_type(8)));

#define DD 256        // feature dim
#define BB 64         // batch
#define TT 2048       // time steps
#define NCHUNK 64     // chunks over the 2047 serial steps
#define CLEN 32       // steps per chunk -> P = (eW^T)^32 via 5 squarings
#define LDSPITCH 260  // row pitch (dwords): row-to-row bank delta 4 -> conflict-free A reads

__device__ __forceinline__ v8f wmma_f32(v2f a, v2f b, v8f c) {
    // D = A(16x4) * B(4x16) + C(16x16), all fp32
    return __builtin_amdgcn_wmma_f32_16x16x4_f32(false, a, false, b, (short)0, c,
                                                 false, false);
}

// ---------------------------------------------------------------- transpose
__global__ void transpose256(const float* __restrict__ src, float* __restrict__ dst) {
    int i = blockIdx.x;   // output row
    int j = threadIdx.x;  // output col (coalesced store)
    dst[i * DD + j] = src[j * DD + i];
}

// ---------------------------------------------------------------- 256^3 GEMM
// C = A @ B row-major; one 16x16 tile per wave, 64 x WMMA K-loop. Matrix squaring.
__global__ __launch_bounds__(256) void gemm256(const float* __restrict__ A,
                                               const float* __restrict__ Bm,
                                               float* __restrict__ C) {
    const int wave = threadIdx.x >> 5;
    const int lane = threadIdx.x & 31;
    const int tile = blockIdx.x * 8 + wave;  // 0..255
    const int m0 = (tile >> 4) << 4;
    const int n0 = (tile & 15) << 4;
    const int m = lane & 15, n = lane & 15, kh = lane >> 4;
    v8f acc = {};
    for (int k0 = 0; k0 < DD; k0 += 4) {
        const int ka = k0 + 2 * kh;
        v2f a = *(const v2f*)(A + (m0 + m) * DD + ka);
        v2f b;
        b.x = Bm[ka * DD + n0 + n];
        b.y = Bm[(ka + 1) * DD + n0 + n];
        acc = wmma_f32(a, b, acc);
    }
    for (int v = 0; v < 8; ++v)
        C[(m0 + v + 8 * kh) * DD + n0 + n] = acc[v];
}

// ---------------------------------------------------------------- chunked scan
// One WG per (chunk, 16-row block); 16 waves, wave w owns column tile n0=16w.
//  - state (16x256) in LDS, single-buffered (two barriers make read->write safe)
//  - I[:,t-1] tile staged in LDS, double-buffered; next tile loaded to regs
//    BEFORE the K-loop so HBM latency overlaps 128 WMMAs, committed at barrier
//  - eW^T B-slice is t-invariant -> held in 128 VGPRs per lane
//  - two accumulator chains (drive vs state) so only the state chain is serial
template <bool WRITE_X>
__global__ __launch_bounds__(512) void scan_chunks(
    const float* __restrict__ x0, const float* __restrict__ I,
    const float* __restrict__ eW, const float* __restrict__ Wie,
    const float* __restrict__ carries,  // [NCHUNK][4][16][256] (WRITE_X)
    float* __restrict__ Vout,           // [NCHUNK][4][16][256] (!WRITE_X)
    float* __restrict__ xout)           // [B][T][D]
{
    __shared__ float st[16][LDSPITCH];
    __shared__ float itile[2][16][LDSPITCH];
    const int chunk = blockIdx.x >> 2;
    const int rblk  = blockIdx.x & 3;
    const int b0 = rblk * 16;
    const int tid = threadIdx.x;
    const int wave = tid >> 5;
    const int lane = tid & 31;
    const int n0 = wave * 16;
    const int m = lane & 15, n = lane & 15, kh = lane >> 4;

    const int tstart = chunk * CLEN + 1;
    const int tlast  = chunk * CLEN + CLEN;
    const int tend   = (tlast < TT - 1) ? tlast : (TT - 1);

    // t-invariant state-transition B operand in registers
    v2f eReg[64];
    const float* Erow = eW + (n0 + n) * DD;
#pragma unroll
    for (int k = 0; k < 64; ++k)
        eReg[k] = *(const v2f*)(Erow + 4 * k + 2 * kh);
    const float* Wrow = Wie + (n0 + n) * DD;  // drive B operand, streamed (L2-resident)

    // init state (zero / carry) + first I tile
    for (int i = tid; i < 16 * DD; i += 512) {
        const int r = i >> 8, c = i & 255;
        float val = 0.f;
        if (WRITE_X) val = carries[((chunk * 4 + rblk) * 16 + r) * DD + c];
        st[r][c] = val;
        itile[0][r][c] = I[((size_t)(b0 + r) * TT + (tstart - 1)) * DD + c];
    }
    if (WRITE_X && chunk == 0) {  // x[:,0] = x0
        for (int i = tid; i < 16 * DD; i += 512) {
            const int r = i >> 8, c = i & 255;
            xout[(size_t)(b0 + r) * TT * DD + c] = x0[(b0 + r) * DD + c];
        }
    }
    __syncthreads();

    int cur = 0;
    for (int t = tstart; t <= tend; ++t) {
        // issue next step's I tile loads early (latency hidden behind WMMAs)
        float pre[8];
        if (t < tend) {
#pragma unroll
            for (int j = 0; j < 8; ++j) {
                const int i = tid + j * 512;
                pre[j] = I[((size_t)(b0 + (i >> 8)) * TT + t) * DD + (i & 255)];
            }
        }
        v8f accd = {};  // drive chain:  I[:,t-1] @ W_inv_eW^T
        v8f accs = {};  // state chain:  y_{t-1} @ eW^T   (serial dependency)
#pragma unroll
        for (int k = 0; k < 64; ++k) {
            const int ka = 4 * k + 2 * kh;
            v2f a1 = *(const v2f*)(&itile[cur][m][ka]);
            v2f b1 = *(const v2f*)(Wrow + ka);
            accd = wmma_f32(a1, b1, accd);
            v2f a2 = *(const v2f*)(&st[m][ka]);
            accs = wmma_f32(a2, eReg[k], accs);
        }
        v8f acc = accd + accs;
        __syncthreads();  // all reads of st / itile[cur] complete
        for (int v = 0; v < 8; ++v)
            st[v + 8 * kh][n0 + n] = acc[v];
        if (t < tend) {
#pragma unroll
            for (int j = 0; j < 8; ++j) {
                const int i = tid + j * 512;
                itile[cur ^ 1][i >> 8][i & 255] = pre[j];
            }
        }
        __syncthreads();
        if (WRITE_X) {
            for (int v = 0; v < 8; ++v)
                xout[((size_t)(b0 + v + 8 * kh) * TT + t) * DD + n0 + n] = acc[v];
        }
        cur ^= 1;
    }
    if (!WRITE_X) {  // emit chunk-final local state V_c
        for (int i = tid; i < 16 * DD; i += 512) {
            const int r = i >> 8, c = i & 255;
            Vout[((chunk * 4 + rblk) * 16 + r) * DD + c] = st[r][c];
        }
    }
}

// ---------------------------------------------------------------- combine
// carry_{c+1} = carry_c @ P + V_c, P = (eW^T)^CLEN held in registers.
__global__ __launch_bounds__(512) void combine_chunks(
    const float* __restrict__ x0, const float* __restrict__ P,
    const float* __restrict__ V, float* __restrict__ carries)
{
    __shared__ float st[16][LDSPITCH];
    const int rblk = blockIdx.x;
    const int b0 = rblk * 16;
    const int tid = threadIdx.x;
    const int wave = tid >> 5, lane = tid & 31;
    const int n0 = wave * 16;
    const int m = lane & 15, n = lane & 15, kh = lane >> 4;

    // chunk-invariant B operand (P) in registers
    v2f pReg[64];
#pragma unroll
    for (int k = 0; k < 64; ++k) {
        const int ka = 4 * k + 2 * kh;
        pReg[k].x = P[ka * DD + n0 + n];
        pReg[k].y = P[(ka + 1) * DD + n0 + n];
    }

    for (int i = tid; i < 16 * DD; i += 512) {
        const int r = i >> 8, c = i & 255;
        const float val = x0[(b0 + r) * DD + c];
        st[r][c] = val;
        carries[(rblk * 16 + r) * DD + c] = val;  // carry_0 = x0
    }
    __syncthreads();

    for (int c = 0; c < NCHUNK - 1; ++c) {
        v8f acc;
        for (int v = 0; v < 8; ++v)
            acc[v] = V[((c * 4 + rblk) * 16 + (v + 8 * kh)) * DD + n0 + n];
#pragma unroll
        for (int k = 0; k < 64; ++k) {
            const int ka = 4 * k + 2 * kh;
            v2f a = *(const v2f*)(&st[m][ka]);
            acc = wmma_f32(a, pReg[k], acc);
        }
        __syncthreads();
        for (int v = 0; v < 8; ++v) {
            st[v + 8 * kh][n0 + n] = acc[v];
            carries[(((c + 1) * 4 + rblk) * 16 + (v + 8 * kh)) * DD + n0 + n] = acc[v];
        }
        __syncthreads();
    }
}

extern "C" void kernel_launch(void* const* d_in, const int* in_sizes, int n_in,
                              void* d_out, int out_size, void* d_ws, size_t ws_size,
                              hipStream_t stream) {
    (void)in_sizes; (void)n_in; (void)out_size; (void)ws_size;
    const float* x0  = (const float*)d_in[0];
    const float* I   = (const float*)d_in[1];
    const float* eW  = (const float*)d_in[2];
    const float* Wie = (const float*)d_in[3];
    float* xout = (float*)d_out;

    float* ws = (float*)d_ws;
    float* p0 = ws;                 // 256x256
    float* p1 = ws + 65536;         // 256x256
    float* V  = ws + 131072;        // NCHUNK*4*16*256 = 1048576 floats
    float* carries = V + 1048576;   // 1048576 floats   (total ~8.9 MB)

    // P = (eW^T)^CLEN via transpose + 5 squarings (log2(32))
    transpose256<<<256, 256, 0, stream>>>(eW, p0);
    const float* src = p0; float* dst = p1;
    for (int i = 0; i < 5; ++i) {
        gemm256<<<32, 256, 0, stream>>>(src, src, dst);
        const float* t = src; src = dst; dst = (float*)t;
    }

    // Pass A: per-chunk local scans from zero state -> V_c
    scan_chunks<false><<<NCHUNK * 4, 512, 0, stream>>>(x0, I, eW, Wie, nullptr, V, nullptr);
    // Combine: 63 serial carry steps (vs 2047 naive)
    combine_chunks<<<4, 512, 0, stream>>>(x0, src, V, carries);
    // Pass B: rerun each chunk from its carry, write x
    scan_chunks<true><<<NCHUNK * 4, 512, 0, stream>>>(x0, I, eW, Wie, carries, nullptr, xout);
}